// WeightedL1NormBackbone_2783138808049
// MI455X (gfx1250) — compile-verified
//
#include <hip/hip_runtime.h>
#include <hip/hip_bf16.h>

// x: [8, 128, 256, 256] f32, weights: [128] f32
// out[b,i,h,w] = (softmax(w)*128)[rank_desc(x[b,:,h,w])[i]] * rsqrt(x^2 + 1e-6)
//
// MI455X reasoning:
//  - 512 MB in+out @ 23.3 TB/s => 22us floor, but ranking needs 8.6G pairwise
//    compares => VALU-bound. Minimize to cmp+carry-add per pair on monotonic
//    u32 keys; 16-way register blocking amortizes LDS reads 16x (ds:valu 1:32).
//  - 128ch x 128px key tile = 64 KB LDS per workgroup (CDNA5 WGP has 320 KB).
//  - CDNA5-specific paths: async global->LDS (ASYNCcnt) for the weight table,
//    global_prefetch_b8 to run the HBM stream ahead. WMMA is inapplicable:
//    the rank indicator 1[x_j > x_i] is not bilinear in any matrix operands.

#define Q        128
#define TILE     128          // pixels per workgroup (= threads per workgroup)
#define SPATIAL  (256 * 256)
#define NB       8
#define EPS_F    1e-6f

// Monotonic DESCENDING key: larger float -> strictly smaller u32. Involution.
__device__ __forceinline__ unsigned desc_key(unsigned u) {
  return u ^ (((int)u >= 0) ? 0x7FFFFFFFu : 0u);
}

__global__ __launch_bounds__(TILE)
void wl1n_rank_kernel(const float* __restrict__ x,
                      const float* __restrict__ w,
                      float* __restrict__ out) {
  __shared__ unsigned keys[Q][TILE];   // 64 KB: transformed keys, [channel][pixel]
  __shared__ float    wbuf[Q];         // raw weights, then wt = softmax(w)*Q

  const int      tid = threadIdx.x;          // one pixel column per thread
  const unsigned wg  = blockIdx.x;
  const unsigned b   = wg / (SPATIAL / TILE);
  const unsigned s0  = (wg % (SPATIAL / TILE)) * TILE;
  const float* xb = x   + (size_t)b * Q * SPATIAL + s0;
  float*       ob = out + (size_t)b * Q * SPATIAL + s0;

  // ---- stage weights: CDNA5 async global->LDS path (ASYNCcnt) ----
#if defined(__gfx1250__) && __has_builtin(__builtin_amdgcn_global_load_async_to_lds_b32)
  {
    typedef __attribute__((address_space(1))) int GInt;   // global (AS1)
    typedef __attribute__((address_space(3))) int LInt;   // LDS (AS3)
    __builtin_amdgcn_global_load_async_to_lds_b32(
        (GInt*)(w + tid), (LInt*)&wbuf[tid], 0, 0);
#if __has_builtin(__builtin_amdgcn_s_wait_asynccnt)
    __builtin_amdgcn_s_wait_asynccnt(0);
#else
    asm volatile("s_wait_asynccnt 0" ::: "memory");
#endif
  }
#else
  wbuf[tid] = w[tid];
#endif
  __syncthreads();

  // ---- softmax(w) * Q, redundant per thread (one-time, branch-free max) ----
  {
    float m = -3.4e38f;
#pragma unroll 8
    for (int q = 0; q < Q; ++q) {
      const float v = wbuf[q];
      m = (v > m) ? v : m;               // compare-select: no NaN scalarization
    }
    float ssum = 0.f;
#pragma unroll 8
    for (int q = 0; q < Q; ++q) ssum += __expf(wbuf[q] - m);
    const float myw = __expf(wbuf[tid] - m) * ((float)Q / ssum);
    __syncthreads();                 // all raw reads done before overwrite
    wbuf[tid] = myw;                 // wbuf now holds wt[0..127]
  }

  // ---- stage x tile: coalesced b128 loads -> key transform -> LDS b128 store ----
#pragma unroll 1
  for (int it = 0; it < Q / 4; ++it) {           // 32 iterations
    const int q = it * 4 + (tid >> 5);           // 4 channel rows per iteration
    const int p = (tid & 31) * 4;                // 4 consecutive pixels
    const float4 v = *(const float4*)(xb + (size_t)q * SPATIAL + p);
    // keep the HBM read stream ahead of the VALU-bound rank phase
    __builtin_prefetch(xb + (size_t)(q + 4) * SPATIAL + p, 0, 0);
    uint4 k;
    k.x = desc_key(__float_as_uint(v.x));
    k.y = desc_key(__float_as_uint(v.y));
    k.z = desc_key(__float_as_uint(v.z));
    k.w = desc_key(__float_as_uint(v.w));
    *(uint4*)&keys[q][p] = k;
  }
  __syncthreads();

  // ---- rank + output: 16 accumulators/thread, 1 LDS read per 16 pairs ----
#pragma unroll 1
  for (int i0 = 0; i0 < Q; i0 += 16) {
    unsigned ki[16];
    int      rk[16];
#pragma unroll
    for (int u = 0; u < 16; ++u) { ki[u] = keys[i0 + u][tid]; rk[u] = 0; }

#pragma unroll 4
    for (int j = 0; j < Q; ++j) {
      const unsigned kj = keys[j][tid];
#pragma unroll
      for (int u = 0; u < 16; ++u) rk[u] += (kj < ki[u]);   // v_cmp + carry-add
    }

#pragma unroll
    for (int u = 0; u < 16; ++u) {
      const float xv = __uint_as_float(desc_key(ki[u]));    // involution -> x
      const float g  = __builtin_amdgcn_rsqf(__builtin_fmaf(xv, xv, EPS_F));
      ob[(size_t)(i0 + u) * SPATIAL] = wbuf[rk[u]] * g;     // LDS gather of wt
    }
  }
}

extern "C" void kernel_launch(void* const* d_in, const int* in_sizes, int n_in,
                              void* d_out, int out_size, void* d_ws, size_t ws_size,
                              hipStream_t stream) {
  const float* x = (const float*)d_in[0];   // 8*128*256*256 f32
  const float* w = (const float*)d_in[1];   // 128 f32
  float* out = (float*)d_out;
  dim3 grid(NB * (SPATIAL / TILE));         // 4096 workgroups
  dim3 block(TILE);                          // 128 threads = 4 wave32
  hipLaunchKernelGGL(wl1n_rank_kernel, grid, block, 0, stream, x, w, out);
  (void)in_sizes; (void)n_in; (void)out_size; (void)d_ws; (void)ws_size;
}